// VecPointNet_14972255994154
// MI455X (gfx1250) — compile-verified
//
#include <hip/hip_runtime.h>
#include <math.h>

#define Nn 1024
#define Hh 128
#define Bb 8
#define Kk 16
#define Ll 4

typedef float v2f __attribute__((ext_vector_type(2)));
typedef float v8f __attribute__((ext_vector_type(8)));

// ---------------------------------------------------------------- zero
__global__ void zero_kernel(float* p, int n) {
  int i = blockIdx.x * blockDim.x + threadIdx.x;
  if (i < n) p[i] = 0.0f;
}

// ---------------------------------------------------------------- KNN (top-16 smallest d2, includes self)
__global__ void knn_kernel(const float* __restrict__ x, int* __restrict__ idx) {
  __shared__ float pts[3 * Nn];
  const int t = threadIdx.x;
  const int b = blockIdx.y;
  const float* xb = x + (long)b * 3 * Nn;
  for (int e = t; e < 3 * Nn; e += 256) pts[e] = xb[e];
  __syncthreads();
  const int n = blockIdx.x * 256 + t;
  const float px = pts[n], py = pts[Nn + n], pz = pts[2 * Nn + n];
  float bd[Kk]; int bi[Kk];
#pragma unroll
  for (int i = 0; i < Kk; ++i) { bd[i] = 3.0e38f; bi[i] = 0; }
  float maxv = 3.0e38f; int maxp = 0;
  for (int m = 0; m < Nn; ++m) {
    const float ddx = pts[m] - px;
    const float ddy = pts[Nn + m] - py;
    const float ddz = pts[2 * Nn + m] - pz;
    const float d2 = ddx * ddx + ddy * ddy + ddz * ddz;
    if (d2 < maxv) {
      bd[maxp] = d2; bi[maxp] = m;
      maxv = bd[0]; maxp = 0;
#pragma unroll
      for (int i = 1; i < Kk; ++i)
        if (bd[i] > maxv) { maxv = bd[i]; maxp = i; }
    }
  }
  int* o = idx + ((long)b * Nn + n) * Kk;
#pragma unroll
  for (int i = 0; i < Kk; ++i) o[i] = bi[i];
}

// ---------------------------------------------------------------- edge feature + first VLNA + mean over K
// One workgroup per (b, n): 16 neighbor slots are the 16 GEMM columns.
__global__ void edge_kernel(const float* __restrict__ x, const int* __restrict__ idx,
                            const float* __restrict__ Wlin,  // [H,3]
                            const float* __restrict__ Wdir,  // [H,H]
                            float* __restrict__ V0) {        // [B,H,3,N]
  __shared__ float feat[3 * 48];     // [ch][xyz][slot]
  __shared__ float qe[Hh * 48];      // q tile: [cin][xyz][slot]
  const int t = threadIdx.x;
  const int n = blockIdx.x, b = blockIdx.y;
  const float* xb = x + (long)b * 3 * Nn;
  const float px = xb[n], py = xb[Nn + n], pz = xb[2 * Nn + n];
  const float pinv = 1.0f / fmaxf(sqrtf(px * px + py * py + pz * pz), 1e-12f);
  const float dxx = px * pinv, dyy = py * pinv, dzz = pz * pinv;
  if (t < Kk) {
    const int m = idx[((long)b * Nn + n) * Kk + t];
    const float nx = xb[m], ny = xb[Nn + m], nz = xb[2 * Nn + m];
    feat[0 * 48 + 0 * 16 + t] = dyy * nz - dzz * ny;   // cross(dir, nbr)
    feat[0 * 48 + 1 * 16 + t] = dzz * nx - dxx * nz;
    feat[0 * 48 + 2 * 16 + t] = dxx * ny - dyy * nx;
    feat[1 * 48 + 0 * 16 + t] = nx - px;               // nbr - x
    feat[1 * 48 + 1 * 16 + t] = ny - py;
    feat[1 * 48 + 2 * 16 + t] = nz - pz;
    feat[2 * 48 + 0 * 16 + t] = px;                    // x
    feat[2 * 48 + 1 * 16 + t] = py;
    feat[2 * 48 + 2 * 16 + t] = pz;
  }
  __syncthreads();
  // q = W_in_lin (Cin=3) x feat
  for (int e = t; e < Hh * 48; e += 256) {
    const int c = e / 48, rem = e - c * 48;
    qe[e] = Wlin[c * 3 + 0] * feat[rem] + Wlin[c * 3 + 1] * feat[48 + rem] +
            Wlin[c * 3 + 2] * feat[96 + rem];
  }
  __syncthreads();
  // k = W_in_dir x q via f32 WMMA: 8 waves * 16 rows, 3 xyz accumulators
  const int lane = t & 31, wv = t >> 5;
  const int j = lane & 15, half = lane >> 4;
  const int arow = wv * 16 + j;
  v8f acc0 = {}, acc1 = {}, acc2 = {};
  for (int kk = 0; kk < Hh; kk += 4) {
    const int c0 = kk + 2 * half;
    v2f a; a.x = Wdir[arow * Hh + c0]; a.y = Wdir[arow * Hh + c0 + 1];
    v2f b0, b1, b2;
    b0.x = qe[c0 * 48 + j];       b0.y = qe[(c0 + 1) * 48 + j];
    b1.x = qe[c0 * 48 + 16 + j];  b1.y = qe[(c0 + 1) * 48 + 16 + j];
    b2.x = qe[c0 * 48 + 32 + j];  b2.y = qe[(c0 + 1) * 48 + 32 + j];
    acc0 = __builtin_amdgcn_wmma_f32_16x16x4_f32(false, a, false, b0, (short)0, acc0, false, false);
    acc1 = __builtin_amdgcn_wmma_f32_16x16x4_f32(false, a, false, b1, (short)0, acc1, false, false);
    acc2 = __builtin_amdgcn_wmma_f32_16x16x4_f32(false, a, false, b2, (short)0, acc2, false, false);
  }
  // VN activation per element, then mean over the 16 neighbor columns
  const float invK = 1.0f / 16.0f;
  float* vb = V0 + (long)b * Hh * 3 * Nn;
#pragma unroll
  for (int r = 0; r < 8; ++r) {
    const int m = wv * 16 + r + 8 * half;
    const float k0 = acc0[r], k1 = acc1[r], k2 = acc2[r];
    const float dnm = fmaxf(sqrtf(k0 * k0 + k1 * k1 + k2 * k2), 1e-12f);
    const float kd0 = k0 / dnm, kd1 = k1 / dnm, kd2 = k2 / dnm;
    const float q0 = qe[m * 48 + j], q1 = qe[m * 48 + 16 + j], q2 = qe[m * 48 + 32 + j];
    const float dt = q0 * kd0 + q1 * kd1 + q2 * kd2;
    const float f = fmaxf(dt, 0.0f) - dt;   // relu(dot)-dot
    float o0 = q0 + f * kd0, o1 = q1 + f * kd1, o2 = q2 + f * kd2;
#pragma unroll
    for (int msk = 1; msk < 16; msk <<= 1) {
      o0 += __shfl_xor(o0, msk, 16);
      o1 += __shfl_xor(o1, msk, 16);
      o2 += __shfl_xor(o2, msk, 16);
    }
    if (j == 0) {
      vb[(m * 3 + 0) * Nn + n] = o0 * invK;
      vb[(m * 3 + 1) * Nn + n] = o1 * invK;
      vb[(m * 3 + 2) * Nn + n] = o2 * invK;
    }
  }
}

// ---------------------------------------------------------------- generic VecLinear GEMM (WMMA f32)
// Y[b,h,xyz,n0+j] = sum_c W[h, c] * X[b,c,xyz,n0+j]  (+ bias[b,h,xyz])
// Optional meanOut: atomically accumulates meanScale * sum_n Y.
__global__ void lin_kernel(const float* __restrict__ W, int ldW, int Kdim,
                           const float* __restrict__ X, long xBS,
                           float* __restrict__ Y, long yBS,
                           const float* __restrict__ bias,
                           float* __restrict__ meanOut, float meanScale) {
  extern __shared__ float tile[];  // [Kdim][3][16]
  const int t = threadIdx.x;
  const int b = blockIdx.y;
  const int n0 = blockIdx.x * 16;
  const float* Xb = X + (long)b * xBS;
  for (int e = t; e < Kdim * 48; e += 256) {
    const int c = e / 48, rem = e - c * 48;
    const int xyz = rem >> 4, nn = rem & 15;
    tile[e] = Xb[(long)(c * 3 + xyz) * Nn + n0 + nn];
  }
  __syncthreads();
  const int lane = t & 31, wv = t >> 5;
  const int j = lane & 15, half = lane >> 4;
  const int row = wv * 16 + j;
  __builtin_prefetch(W + (long)row * ldW, 0, 3);
  v8f acc0 = {}, acc1 = {}, acc2 = {};
  for (int kk = 0; kk < Kdim; kk += 4) {
    const int c0 = kk + 2 * half;
    v2f a; a.x = W[(long)row * ldW + c0]; a.y = W[(long)row * ldW + c0 + 1];
    v2f b0, b1, b2;
    b0.x = tile[c0 * 48 + j];       b0.y = tile[(c0 + 1) * 48 + j];
    b1.x = tile[c0 * 48 + 16 + j];  b1.y = tile[(c0 + 1) * 48 + 16 + j];
    b2.x = tile[c0 * 48 + 32 + j];  b2.y = tile[(c0 + 1) * 48 + 32 + j];
    acc0 = __builtin_amdgcn_wmma_f32_16x16x4_f32(false, a, false, b0, (short)0, acc0, false, false);
    acc1 = __builtin_amdgcn_wmma_f32_16x16x4_f32(false, a, false, b1, (short)0, acc1, false, false);
    acc2 = __builtin_amdgcn_wmma_f32_16x16x4_f32(false, a, false, b2, (short)0, acc2, false, false);
  }
  float* Yb = Y + (long)b * yBS;
#pragma unroll
  for (int r = 0; r < 8; ++r) {
    const int m = wv * 16 + r + 8 * half;
    float o0 = acc0[r], o1 = acc1[r], o2 = acc2[r];
    if (bias) {
      o0 += bias[(b * Hh + m) * 3 + 0];
      o1 += bias[(b * Hh + m) * 3 + 1];
      o2 += bias[(b * Hh + m) * 3 + 2];
    }
    Yb[(long)(m * 3 + 0) * Nn + n0 + j] = o0;
    Yb[(long)(m * 3 + 1) * Nn + n0 + j] = o1;
    Yb[(long)(m * 3 + 2) * Nn + n0 + j] = o2;
    if (meanOut) {
      float s0 = o0, s1 = o1, s2 = o2;
#pragma unroll
      for (int msk = 1; msk < 16; msk <<= 1) {
        s0 += __shfl_xor(s0, msk, 16);
        s1 += __shfl_xor(s1, msk, 16);
        s2 += __shfl_xor(s2, msk, 16);
      }
      if (j == 0) {
        atomicAdd(&meanOut[(b * Hh + m) * 3 + 0], s0 * meanScale);
        atomicAdd(&meanOut[(b * Hh + m) * 3 + 1], s1 * meanScale);
        atomicAdd(&meanOut[(b * Hh + m) * 3 + 2], s2 * meanScale);
      }
    }
  }
}

// ---------------------------------------------------------------- k = Wd x q, then VN activation (fused)
__global__ void act_kernel(const float* __restrict__ Wd,   // [H,H]
                           const float* __restrict__ Q,    // [B,H,3,N], bstride H*3*N
                           float* __restrict__ Y, long yBS) {
  __shared__ float tile[Hh * 48];
  const int t = threadIdx.x;
  const int b = blockIdx.y;
  const int n0 = blockIdx.x * 16;
  const float* Qb = Q + (long)b * Hh * 3 * Nn;
  for (int e = t; e < Hh * 48; e += 256) {
    const int c = e / 48, rem = e - c * 48;
    const int xyz = rem >> 4, nn = rem & 15;
    tile[e] = Qb[(long)(c * 3 + xyz) * Nn + n0 + nn];
  }
  __syncthreads();
  const int lane = t & 31, wv = t >> 5;
  const int j = lane & 15, half = lane >> 4;
  const int row = wv * 16 + j;
  v8f acc0 = {}, acc1 = {}, acc2 = {};
  for (int kk = 0; kk < Hh; kk += 4) {
    const int c0 = kk + 2 * half;
    v2f a; a.x = Wd[row * Hh + c0]; a.y = Wd[row * Hh + c0 + 1];
    v2f b0, b1, b2;
    b0.x = tile[c0 * 48 + j];       b0.y = tile[(c0 + 1) * 48 + j];
    b1.x = tile[c0 * 48 + 16 + j];  b1.y = tile[(c0 + 1) * 48 + 16 + j];
    b2.x = tile[c0 * 48 + 32 + j];  b2.y = tile[(c0 + 1) * 48 + 32 + j];
    acc0 = __builtin_amdgcn_wmma_f32_16x16x4_f32(false, a, false, b0, (short)0, acc0, false, false);
    acc1 = __builtin_amdgcn_wmma_f32_16x16x4_f32(false, a, false, b1, (short)0, acc1, false, false);
    acc2 = __builtin_amdgcn_wmma_f32_16x16x4_f32(false, a, false, b2, (short)0, acc2, false, false);
  }
  float* Yb = Y + (long)b * yBS;
#pragma unroll
  for (int r = 0; r < 8; ++r) {
    const int m = wv * 16 + r + 8 * half;
    const float k0 = acc0[r], k1 = acc1[r], k2 = acc2[r];
    const float dnm = fmaxf(sqrtf(k0 * k0 + k1 * k1 + k2 * k2), 1e-12f);
    const float kd0 = k0 / dnm, kd1 = k1 / dnm, kd2 = k2 / dnm;
    const float q0 = tile[m * 48 + j], q1 = tile[m * 48 + 16 + j], q2 = tile[m * 48 + 32 + j];
    const float dt = q0 * kd0 + q1 * kd1 + q2 * kd2;
    const float f = fmaxf(dt, 0.0f) - dt;
    Yb[(long)(m * 3 + 0) * Nn + n0 + j] = q0 + f * kd0;
    Yb[(long)(m * 3 + 1) * Nn + n0 + j] = q1 + f * kd1;
    Yb[(long)(m * 3 + 2) * Nn + n0 + j] = q2 + f * kd2;
  }
}

// ---------------------------------------------------------------- global mean pool over N
__global__ void pool_kernel(const float* __restrict__ Y, float* __restrict__ yg) {
  __shared__ float red[256];
  const int t = threadIdx.x;
  const int h = blockIdx.x, b = blockIdx.y;
  const float* Yb = Y + ((long)b * Hh + h) * 3 * Nn;
  for (int xyz = 0; xyz < 3; ++xyz) {
    float s = 0.0f;
    for (int n = t; n < Nn; n += 256) s += Yb[(long)xyz * Nn + n];
    red[t] = s;
    __syncthreads();
    for (int off = 128; off > 0; off >>= 1) {
      if (t < off) red[t] += red[t + off];
      __syncthreads();
    }
    if (t == 0) yg[(b * Hh + h) * 3 + xyz] = red[0] / (float)Nn;
    __syncthreads();
  }
}

// ---------------------------------------------------------------- fold concat'd global half into a bias
// bias[b,h,xyz] = sum_c Wg[h, H + c] * yg[b,c,xyz]
__global__ void bias_kernel(const float* __restrict__ Wg, const float* __restrict__ yg,
                            float* __restrict__ bias) {
  const int h = threadIdx.x;
  const int xyz = blockIdx.x, b = blockIdx.y;
  float s = 0.0f;
  for (int c = 0; c < Hh; ++c)
    s += Wg[h * (2 * Hh) + Hh + c] * yg[(b * Hh + c) * 3 + xyz];
  bias[(b * Hh + h) * 3 + xyz] = s;
}

// ---------------------------------------------------------------- launch
extern "C" void kernel_launch(void* const* d_in, const int* in_sizes, int n_in,
                              void* d_out, int out_size, void* d_ws, size_t ws_size,
                              hipStream_t stream) {
  const float* x         = (const float*)d_in[0];
  const float* w_in_lin  = (const float*)d_in[1];
  const float* w_in_dir  = (const float*)d_in[2];
  const float* w_lay_lin = (const float*)d_in[3];
  const float* w_lay_dir = (const float*)d_in[4];
  const float* w_glb_lin = (const float*)d_in[5];
  const float* w_glb_dir = (const float*)d_in[6];
  const float* w_out     = (const float*)d_in[7];
  float* out = (float*)d_out;

  const long H3N = (long)Hh * 3 * Nn;   // 393216
  const long F3N = (long)Ll * H3N;      // 1572864

  // workspace layout (floats)
  int*   idx   = (int*)d_ws;
  float* base  = (float*)d_ws;
  float* v0    = base + (long)Bb * Nn * Kk;       // [B,H,3,N]
  float* tmp1  = v0   + (long)Bb * H3N;           // q scratch
  float* tmp2  = tmp1 + (long)Bb * H3N;           // y1 scratch
  float* yg    = tmp2 + (long)Bb * H3N;           // [B,H,3]
  float* bias  = yg   + (long)Bb * Hh * 3;        // [B,H,3]
  float* feats = bias + (long)Bb * Hh * 3;        // [B,4H,3,N]

  // zero the mean-output accumulator region of d_out
  zero_kernel<<<dim3((Bb * Hh * 3 + 255) / 256), 256, 0, stream>>>(out, Bb * Hh * 3);

  knn_kernel<<<dim3(Nn / 256, Bb), 256, 0, stream>>>(x, idx);
  edge_kernel<<<dim3(Nn, Bb), 256, 0, stream>>>(x, idx, w_in_lin, w_in_dir, v0);

  for (int i = 0; i < Ll; ++i) {
    const float* inP = (i == 0) ? v0 : (feats + (long)(i - 1) * H3N);
    const long inBS = (i == 0) ? H3N : F3N;
    // q = W_lay_lin x y
    lin_kernel<<<dim3(Nn / 16, Bb), 256, Hh * 48 * sizeof(float), stream>>>(
        w_lay_lin + (long)i * Hh * Hh, Hh, Hh, inP, inBS, tmp1, H3N,
        nullptr, nullptr, 0.0f);
    // y1 = VN-act(q, W_lay_dir)
    act_kernel<<<dim3(Nn / 16, Bb), 256, 0, stream>>>(
        w_lay_dir + (long)i * Hh * Hh, tmp1, tmp2, H3N);
    // yg = mean_N(y1); bias = W_glb_lin[:,H:] x yg
    pool_kernel<<<dim3(Hh, Bb), 256, 0, stream>>>(tmp2, yg);
    bias_kernel<<<dim3(3, Bb), Hh, 0, stream>>>(
        w_glb_lin + (long)i * Hh * 2 * Hh, yg, bias);
    // q2 = W_glb_lin[:,:H] x y1 + bias
    lin_kernel<<<dim3(Nn / 16, Bb), 256, Hh * 48 * sizeof(float), stream>>>(
        w_glb_lin + (long)i * Hh * 2 * Hh, 2 * Hh, Hh, tmp2, H3N, tmp1, H3N,
        bias, nullptr, 0.0f);
    // y = VN-act(q2, W_glb_dir) -> written directly into feats slice i
    act_kernel<<<dim3(Nn / 16, Bb), 256, 0, stream>>>(
        w_glb_dir + (long)i * Hh * Hh, tmp1, feats + (long)i * H3N, F3N);
  }

  // out = W_out[128,512] x feats; fused mean over N into d_out[0 .. B*C*3)
  lin_kernel<<<dim3(Nn / 16, Bb), 256, 4 * Hh * 48 * sizeof(float), stream>>>(
      w_out, 4 * Hh, 4 * Hh, feats, F3N, out + (long)Bb * Hh * 3, (long)Hh * 3 * Nn,
      nullptr, out, 1.0f / (float)Nn);

  (void)in_sizes; (void)n_in; (void)out_size; (void)ws_size;
}